// RTFFSSM_Block_24292335026642
// MI455X (gfx1250) — compile-verified
//
#include <hip/hip_runtime.h>
#include <hip/hip_bf16.h>
#include <math.h>

// Sizes fixed by the reference
#define B_SZ  8
#define SEQ   8192
#define CH    256
#define ORDER 64

typedef float v2f __attribute__((ext_vector_type(2)));
typedef float v8f __attribute__((ext_vector_type(8)));

__device__ __forceinline__ float gelu_exact(float x) {
    return 0.5f * x * (1.0f + erff(x * 0.70710678118654752f));
}

// ---------------------------------------------------------------------------
// WMMA fragment loaders, V_WMMA_F32_16X16X4_F32 layouts (ISA 7.12.2):
//  A 16x4 : lanes 0-15 -> M=lane, VGPR0=K0,VGPR1=K1 ; lanes16-31 -> K2,K3
//  B 4x16 : lanes 0-15 -> N=lane, VGPR0=K0,VGPR1=K1 ; lanes16-31 -> K2,K3
//  C 16x16: lanes 0-15 -> M=g,N=lane ; lanes16-31 -> M=8+g,N=lane-16
// ---------------------------------------------------------------------------
__device__ __forceinline__ v2f load_a_rowmajor(const float* __restrict__ A,
                                               int mbase, int mt, int k0, int lane) {
    int m  = mbase + mt * 16 + (lane & 15);
    int kk = k0 + ((lane >> 4) << 1);
    const float* p = A + (size_t)m * CH + kk;
    v2f a; a.x = p[0]; a.y = p[1];
    return a;
}
// A element (m, k) taken from channel-major buffer yT[(b*CH + k)*SEQ + t]
__device__ __forceinline__ v2f load_a_chmajor(const float* __restrict__ yT,
                                              int mbase, int mt, int k0, int lane) {
    int m  = mbase + mt * 16 + (lane & 15);
    int kk = k0 + ((lane >> 4) << 1);
    int b  = m >> 13;            // SEQ = 8192
    int t  = m & (SEQ - 1);
    const float* p = yT + ((size_t)(b * CH + kk) * SEQ + t);
    v2f a; a.x = p[0]; a.y = p[SEQ];
    return a;
}
// B[k][n] = W[n][k]  (computing X @ W^T), W row-major (CH, CH)
__device__ __forceinline__ v2f load_b_wT(const float* __restrict__ W,
                                         int n0, int k0, int lane) {
    int n  = n0 + (lane & 15);
    int kk = k0 + ((lane >> 4) << 1);
    const float* p = W + (size_t)n * CH + kk;
    v2f b; b.x = p[0]; b.y = p[1];
    return b;
}

// ---------------------------------------------------------------------------
// K0: zero the max-abs stat buffers (harness poisons ws; atomics need 0 init)
// ---------------------------------------------------------------------------
__global__ void k_zero_stats(unsigned* __restrict__ stats) {
    int i = blockIdx.x * 256 + threadIdx.x;
    if (i < 2 * B_SZ * CH) stats[i] = 0u;
}

// ---------------------------------------------------------------------------
// K1: xe = x @ W_enc^T + b_enc ; also per-(b,c) max|xe| via atomics
// grid = 65536/64 blocks of 256 threads. Wave w: M-tile (w&3), 8 N-tiles.
// ---------------------------------------------------------------------------
__global__ __launch_bounds__(256) void k_gemm_enc(
        const float* __restrict__ x, const float* __restrict__ W,
        const float* __restrict__ bias, float* __restrict__ xe,
        unsigned* __restrict__ xemax) {
    __shared__ unsigned smax[CH];
    const int tid = threadIdx.x, lane = tid & 31, w = tid >> 5;
    const int mbase = blockIdx.x * 64;
    const int mt = w & 3, nhalf = w >> 2;
    smax[tid] = 0u;
    __syncthreads();

    v8f zero = {0.f, 0.f, 0.f, 0.f, 0.f, 0.f, 0.f, 0.f};
    v8f acc[8];
#pragma unroll
    for (int j = 0; j < 8; ++j) acc[j] = zero;

    for (int k0 = 0; k0 < CH; k0 += 4) {
        v2f a = load_a_rowmajor(x, mbase, mt, k0, lane);
#pragma unroll
        for (int j = 0; j < 8; ++j) {
            int n0 = (nhalf * 8 + j) * 16;
            v2f bf = load_b_wT(W, n0, k0, lane);
            acc[j] = __builtin_amdgcn_wmma_f32_16x16x4_f32(
                false, a, false, bf, (short)0, acc[j], false, false);
        }
    }
    const int mrow0 = mbase + mt * 16 + ((lane >> 4) ? 8 : 0);
#pragma unroll
    for (int j = 0; j < 8; ++j) {
        int col = (nhalf * 8 + j) * 16 + (lane & 15);
        float bn = bias[col];
        float lmax = 0.f;
#pragma unroll
        for (int g = 0; g < 8; ++g) {
            float v = acc[j][g] + bn;
            xe[(size_t)(mrow0 + g) * CH + col] = v;
            lmax = fmaxf(lmax, fabsf(v));
        }
        atomicMax(&smax[col], __float_as_uint(lmax));   // nonneg floats: bit order ok
    }
    __syncthreads();
    int b = mbase >> 13;
    atomicMax(&xemax[b * CH + tid], smax[tid]);
}

// ---------------------------------------------------------------------------
// K2: order-64 IIR recurrence == causal conv with truncated B(z)/A(z) impulse
// response (zero initial state).  One wave per (b, c).  WG = 32 waves = one
// batch row, 32 channels; cooperative coalesced staging of 32x32 x-tiles.
// Writes y1 = gelu(h1 + h0*xe) transposed to (B, C, L) for coalesced stores.
// ---------------------------------------------------------------------------
__global__ __launch_bounds__(1024) void k_iir(
        const float* __restrict__ xe, const float* __restrict__ A,
        const float* __restrict__ Bp, const float* __restrict__ h0p,
        float* __restrict__ y1T) {
    __shared__ float xr[32 * 129];   // x ring, depth 128, pad 129 (conflict-free)
    __shared__ float ur[32 * 65];    // u ring, depth 64,  pad 65
    const int tid = threadIdx.x, lane = tid & 31, w = tid >> 5;
    const int c0 = blockIdx.x * 32, b = blockIdx.y;
    const int c = c0 + w;

    for (int i = tid; i < 32 * 129; i += 1024) xr[i] = 0.f;
    for (int i = tid; i < 32 * 65;  i += 1024) ur[i] = 0.f;

    // lane l covers FIR taps j=l, j=l+32 and feedback taps j=l+1, j=l+33
    const float a1 = A[c * ORDER + lane];
    const float a2 = A[c * ORDER + 32 + lane];
    const float b1 = Bp[c * ORDER + lane];
    const float b2 = Bp[c * ORDER + 32 + lane];
    const float h0 = h0p[0];

    const float* xrow = xe + (size_t)b * SEQ * CH + c0;       // [t*CH + ch]
    float* yout = y1T + (size_t)(b * CH + c) * SEQ;
    const float* xrc = xr + w * 129;
    const float* urc = ur + w * 65;
    float* urw = ur + w * 65;
    float keep = 0.f;

    for (int t0 = 0; t0 < SEQ; t0 += 32) {
        float xv = xrow[(size_t)(t0 + w) * CH + lane];        // wave w: row t0+w
        if (t0 + 32 < SEQ)
            __builtin_prefetch(&xrow[(size_t)(t0 + 32 + w) * CH + lane], 0, 0);
        __syncthreads();                                      // prior reads done
        xr[lane * 129 + ((t0 + w) & 127)] = xv;               // ch=lane, time t0+w
        __syncthreads();                                      // staging visible
#pragma unroll 4
        for (int dt = 0; dt < 32; ++dt) {
            int t = t0 + dt;
            float s = b1 * xrc[(t - lane) & 127]
                    + b2 * xrc[(t - 32 - lane) & 127]
                    - a1 * urc[(t - 1 - lane) & 63]
                    - a2 * urc[(t - 33 - lane) & 63];
#pragma unroll
            for (int o = 16; o; o >>= 1) s += __shfl_xor(s, o, 32);
            if (lane == 0) urw[t & 63] = s;                   // in-order DS per wave
            float yv = gelu_exact(s + h0 * xrc[t & 127]);
            if (lane == dt) keep = yv;
        }
        yout[t0 + lane] = keep;                               // coalesced 128B
    }
}

// ---------------------------------------------------------------------------
// K3: y2 = gelu(y1 @ W_fc^T + b_fc) + xe   (in-place over xe buffer)
// ---------------------------------------------------------------------------
__global__ __launch_bounds__(256) void k_gemm_fc(
        const float* __restrict__ y1T, const float* __restrict__ W,
        const float* __restrict__ bias, float* xe /* read skip + write y2 */) {
    const int tid = threadIdx.x, lane = tid & 31, w = tid >> 5;
    const int mbase = blockIdx.x * 64;
    const int mt = w & 3, nhalf = w >> 2;

    v8f zero = {0.f, 0.f, 0.f, 0.f, 0.f, 0.f, 0.f, 0.f};
    v8f acc[8];
#pragma unroll
    for (int j = 0; j < 8; ++j) acc[j] = zero;

    for (int k0 = 0; k0 < CH; k0 += 4) {
        v2f a = load_a_chmajor(y1T, mbase, mt, k0, lane);
#pragma unroll
        for (int j = 0; j < 8; ++j) {
            int n0 = (nhalf * 8 + j) * 16;
            v2f bf = load_b_wT(W, n0, k0, lane);
            acc[j] = __builtin_amdgcn_wmma_f32_16x16x4_f32(
                false, a, false, bf, (short)0, acc[j], false, false);
        }
    }
    const int mrow0 = mbase + mt * 16 + ((lane >> 4) ? 8 : 0);
#pragma unroll
    for (int j = 0; j < 8; ++j) {
        int col = (nhalf * 8 + j) * 16 + (lane & 15);
        float bn = bias[col];
#pragma unroll
        for (int g = 0; g < 8; ++g) {
            size_t idx = (size_t)(mrow0 + g) * CH + col;
            xe[idx] = gelu_exact(acc[j][g] + bn) + xe[idx];
        }
    }
}

// ---------------------------------------------------------------------------
// K4: LayerNorm over channels, in place.  One wave per (b,t) row.
// ---------------------------------------------------------------------------
__global__ __launch_bounds__(256) void k_ln(
        float* __restrict__ y2, const float* __restrict__ gamma,
        const float* __restrict__ beta) {
    const int lane = threadIdx.x & 31, w = threadIdx.x >> 5;
    const size_t row = (size_t)blockIdx.x * 8 + w;
    float* p = y2 + row * CH;
    float v[8], sum = 0.f;
#pragma unroll
    for (int j = 0; j < 8; ++j) { v[j] = p[lane + 32 * j]; sum += v[j]; }
#pragma unroll
    for (int o = 16; o; o >>= 1) sum += __shfl_xor(sum, o, 32);
    float mu = sum * (1.f / CH);
    float ss = 0.f;
#pragma unroll
    for (int j = 0; j < 8; ++j) { float d = v[j] - mu; ss += d * d; }
#pragma unroll
    for (int o = 16; o; o >>= 1) ss += __shfl_xor(ss, o, 32);
    float rs = rsqrtf(ss * (1.f / CH) + 1e-5f);
#pragma unroll
    for (int j = 0; j < 8; ++j) {
        int cidx = lane + 32 * j;
        p[cidx] = (v[j] - mu) * rs * gamma[cidx] + beta[cidx];
    }
}

// ---------------------------------------------------------------------------
// K5: y = y2 @ W_dec^T + b_dec -> d_out ; per-(b,c) max|y| via atomics
// ---------------------------------------------------------------------------
__global__ __launch_bounds__(256) void k_gemm_dec(
        const float* __restrict__ y2, const float* __restrict__ W,
        const float* __restrict__ bias, float* __restrict__ y,
        unsigned* __restrict__ ymax) {
    __shared__ unsigned smax[CH];
    const int tid = threadIdx.x, lane = tid & 31, w = tid >> 5;
    const int mbase = blockIdx.x * 64;
    const int mt = w & 3, nhalf = w >> 2;
    smax[tid] = 0u;
    __syncthreads();

    v8f zero = {0.f, 0.f, 0.f, 0.f, 0.f, 0.f, 0.f, 0.f};
    v8f acc[8];
#pragma unroll
    for (int j = 0; j < 8; ++j) acc[j] = zero;

    for (int k0 = 0; k0 < CH; k0 += 4) {
        v2f a = load_a_rowmajor(y2, mbase, mt, k0, lane);
#pragma unroll
        for (int j = 0; j < 8; ++j) {
            int n0 = (nhalf * 8 + j) * 16;
            v2f bf = load_b_wT(W, n0, k0, lane);
            acc[j] = __builtin_amdgcn_wmma_f32_16x16x4_f32(
                false, a, false, bf, (short)0, acc[j], false, false);
        }
    }
    const int mrow0 = mbase + mt * 16 + ((lane >> 4) ? 8 : 0);
#pragma unroll
    for (int j = 0; j < 8; ++j) {
        int col = (nhalf * 8 + j) * 16 + (lane & 15);
        float bn = bias[col];
        float lmax = 0.f;
#pragma unroll
        for (int g = 0; g < 8; ++g) {
            float v = acc[j][g] + bn;
            y[(size_t)(mrow0 + g) * CH + col] = v;
            lmax = fmaxf(lmax, fabsf(v));
        }
        atomicMax(&smax[col], __float_as_uint(lmax));
    }
    __syncthreads();
    int b = mbase >> 13;
    atomicMax(&ymax[b * CH + tid], smax[tid]);
}

// ---------------------------------------------------------------------------
// K6: h = max|y| / (max|xe| + 1e-6)
// ---------------------------------------------------------------------------
__global__ void k_hout(const unsigned* __restrict__ xemax,
                       const unsigned* __restrict__ ymax,
                       float* __restrict__ hout) {
    int i = blockIdx.x * 256 + threadIdx.x;
    if (i < B_SZ * CH)
        hout[i] = __uint_as_float(ymax[i]) / (__uint_as_float(xemax[i]) + 1e-6f);
}

// ---------------------------------------------------------------------------
extern "C" void kernel_launch(void* const* d_in, const int* in_sizes, int n_in,
                              void* d_out, int out_size, void* d_ws, size_t ws_size,
                              hipStream_t stream) {
    (void)in_sizes; (void)n_in; (void)out_size; (void)ws_size;
    const float* x     = (const float*)d_in[0];
    const float* A     = (const float*)d_in[1];
    const float* Bp    = (const float*)d_in[2];
    const float* h0    = (const float*)d_in[3];
    const float* W_enc = (const float*)d_in[4];
    const float* b_enc = (const float*)d_in[5];
    const float* W_fc  = (const float*)d_in[6];
    const float* b_fc  = (const float*)d_in[7];
    const float* gamma = (const float*)d_in[8];
    const float* beta  = (const float*)d_in[9];
    const float* W_dec = (const float*)d_in[10];
    const float* b_dec = (const float*)d_in[11];

    float* y    = (float*)d_out;                       // (B, L, C)
    float* hout = y + (size_t)B_SZ * SEQ * CH;         // (B, C)

    char* ws = (char*)d_ws;                            // needs 128 MiB + 16 KiB
    float*    xe    = (float*)ws;                                   // xe, then y2 in place
    float*    y1T   = (float*)(ws + (size_t)64 * 1024 * 1024);      // (B, C, L)
    unsigned* xemax = (unsigned*)(ws + (size_t)128 * 1024 * 1024);
    unsigned* ymax  = xemax + B_SZ * CH;

    const int mblocks = (B_SZ * SEQ) / 64;             // 1024

    k_zero_stats<<<(2 * B_SZ * CH + 255) / 256, 256, 0, stream>>>(xemax);
    k_gemm_enc<<<mblocks, 256, 0, stream>>>(x, W_enc, b_enc, xe, xemax);
    dim3 g2(CH / 32, B_SZ);
    k_iir<<<g2, 1024, 0, stream>>>(xe, A, Bp, h0, y1T);
    k_gemm_fc<<<mblocks, 256, 0, stream>>>(y1T, W_fc, b_fc, xe);
    k_ln<<<(B_SZ * SEQ) / 8, 256, 0, stream>>>(xe, gamma, beta);
    k_gemm_dec<<<mblocks, 256, 0, stream>>>(xe, W_dec, b_dec, y, ymax);
    k_hout<<<(B_SZ * CH + 255) / 256, 256, 0, stream>>>(xemax, ymax, hout);
}